// MemoryMLP_65575560675770
// MI455X (gfx1250) — compile-verified
//
#include <hip/hip_runtime.h>
#include <math.h>

// ============================================================================
// MemoryMLP (TTT 2-layer MLP memory) for MI455X / gfx1250.
//
// Design (from MI455X specs):
//  * 75.5 MB of recurrent state (W1,SW1,W2,SW2 per batch) fits in the 192 MB
//    global L2 -> keep it L2-resident, stream ~26 MB/step/batch through VALU.
//  * Persistent cooperative kernel: 8 batches x 24 workgroups (192 WGs of 256
//    threads).  Each WG owns a 32-wide h (==d) slice of the state; per-batch
//    sense-counter barriers (3 per step) order the phases.
//  * Matvecs (z1, z2, dh) use V_WMMA_F32_16X16X4_F32 with the vector
//    broadcast into A (f32 end-to-end, 64 MACs/instr vs 32 for v_fmac).
//  * z1 / dh / delta1 / b1 / b2 slices live in LDS for the whole scan; only
//    h1 and delta2 (768 floats each) cross workgroups, through L2.
// ============================================================================

typedef __attribute__((ext_vector_type(2))) float v2f;
typedef __attribute__((ext_vector_type(8))) float v8f;

#define GC 0.7978845608028654f
#define GA 0.044715f

constexpr int kB = 8, kT = 256, kD = 768, kH = 768;
constexpr int kG = 24;         // workgroups per batch
constexpr int kS = 32;         // slice width per workgroup (kG * kS == 768)
constexpr int kThreads = 256;  // 8 waves (wave32)

__device__ __forceinline__ float gelu_fwd(float x) {
  float t = tanhf(GC * (x + GA * x * x * x));
  return 0.5f * x * (1.f + t);
}
__device__ __forceinline__ float gelu_bwd(float x) {
  float t = tanhf(GC * (x + GA * x * x * x));
  return 0.5f * (1.f + t) + 0.5f * x * (1.f - t * t) * GC * (1.f + 3.f * GA * x * x);
}

__device__ __forceinline__ v8f wmma_f32_16x16x4(v2f a, v2f b, v8f c) {
  // args: (neg_a, A, neg_b, B, c_mod, C, reuse_a, reuse_b)
  return __builtin_amdgcn_wmma_f32_16x16x4_f32(false, a, false, b, (short)0, c,
                                               false, false);
}

// Per-batch barrier over kG workgroups: monotone counter, target = epoch*kG.
__device__ __forceinline__ void batch_barrier(unsigned* bar, unsigned target) {
  __syncthreads();
  if (threadIdx.x == 0) {
    __threadfence();  // make this WG's global writes visible (agent scope)
    __hip_atomic_fetch_add(bar, 1u, __ATOMIC_RELEASE, __HIP_MEMORY_SCOPE_AGENT);
    while (__hip_atomic_load(bar, __ATOMIC_ACQUIRE, __HIP_MEMORY_SCOPE_AGENT) <
           target) {
      __builtin_amdgcn_s_sleep(1);
    }
  }
  __syncthreads();
}

__global__ void __launch_bounds__(kThreads)
memory_mlp_kernel(const float* __restrict__ K, const float* __restrict__ V,
                  const float* __restrict__ ETA, const float* __restrict__ ALPHA,
                  const float* __restrict__ THETA,
                  const float* __restrict__ W01, const float* __restrict__ B01,
                  const float* __restrict__ W02, const float* __restrict__ B02,
                  float* __restrict__ OUT, float* __restrict__ WS,
                  unsigned* __restrict__ BARS) {
  const int blk = blockIdx.x;
  const int b = blk / kG;
  const int sl = blk % kG;
  const int h0 = sl * kS;  // owned h-slice; same index used as d-slice for z2/b2
  const int tid = threadIdx.x;
  const int lane = tid & 31;
  const int wave = tid >> 5;

  const size_t MAT = (size_t)kD * kH;
  float* W1 = WS + (size_t)b * (4 * MAT + 2 * (size_t)kH);
  float* SW1 = W1 + MAT;
  float* W2 = SW1 + MAT;
  float* SW2 = W2 + MAT;
  float* H1G = SW2 + MAT;  // shared h1 vector (768)
  float* D2G = H1G + kH;   // shared delta2 vector (768)
  unsigned* bar = BARS + b;

  __shared__ float k_lds[kD];
  __shared__ float h1_lds[kH];
  __shared__ float d2_lds[kD];
  __shared__ float z1_lds[kS];
  __shared__ float dh_lds[kS];
  __shared__ float dl1_lds[kS];
  __shared__ float b1_lds[kS], Sb1_lds[kS], b2_lds[kS], Sb2_lds[kS];
  __shared__ float part_lds[2][4][16];  // WMMA partial sums [group][Kquarter][n]

  // ---- state init (rebuilt every launch: deterministic) ----
  for (int d = wave; d < kD; d += 8) {  // W1 columns [all d][h0..h0+32)
    size_t idx = (size_t)d * kH + h0 + lane;
    W1[idx] = W01[idx];
    SW1[idx] = 0.f;
  }
  for (int r = wave; r < kS; r += 8) {  // W2 rows [h0..h0+32)[all d]
    size_t base = (size_t)(h0 + r) * kD;
    for (int d = lane; d < kD; d += 32) {
      W2[base + d] = W02[base + d];
      SW2[base + d] = 0.f;
    }
  }
  if (tid < kS) {
    b1_lds[tid] = B01[h0 + tid];
    Sb1_lds[tid] = 0.f;
    b2_lds[tid] = B02[h0 + tid];
    Sb2_lds[tid] = 0.f;
  }
  __syncthreads();

  unsigned ep = 0;
  for (int t = 0; t < kT; ++t) {
    const float et = ETA[b * kT + t];
    const float at = ALPHA[b * kT + t];
    const float th = THETA[b * kT + t];
    const float oma = 1.f - at;
    const float* kt = K + ((size_t)b * kT + t) * (size_t)kD;
    const float* vt = V + ((size_t)b * kT + t) * (size_t)kD;

    // stage k_t (used by P1 WMMA A-operand and by the W1 update)
    for (int i = tid; i < kD; i += kThreads) k_lds[i] = kt[i];
    __threadfence_block();  // our own W1/W2 updates from t-1 -> visible in-WG
    __syncthreads();

    // ---- P1: z1[slice] = k . W1[:,slice]  (WMMA, K split 4 ways) ----
    {
      const int hg = wave & 1;   // 16-wide h group within slice
      const int dq = wave >> 1;  // 192-long K-range quarter
      const int n = lane & 15;
      const int kk = (lane < 16) ? 0 : 2;
      const float* bcol = W1 + (size_t)kk * kH + h0 + hg * 16 + n;
      v8f acc0 = {0.f, 0.f, 0.f, 0.f, 0.f, 0.f, 0.f, 0.f};
      v8f acc1 = acc0;
      const int dbeg = dq * 192;
      for (int dd = 0; dd < 192; dd += 8) {
        const int d = dbeg + dd;
        v2f a0, a1, bb0, bb1;
        a0.x = k_lds[d + kk];
        a0.y = k_lds[d + kk + 1];
        a1.x = k_lds[d + 4 + kk];
        a1.y = k_lds[d + 4 + kk + 1];
        const float* p0 = bcol + (size_t)d * kH;
        const float* p1 = p0 + (size_t)4 * kH;
        bb0.x = p0[0];
        bb0.y = p0[kH];
        bb1.x = p1[0];
        bb1.y = p1[kH];
        acc0 = wmma_f32_16x16x4(a0, bb0, acc0);
        acc1 = wmma_f32_16x16x4(a1, bb1, acc1);
      }
      v8f acc = acc0 + acc1;
      if (lane < 16) part_lds[hg][dq][n] = acc[0];  // M=0 row (all rows equal)
    }
    __syncthreads();
    if (tid < kS) {
      const int g = tid >> 4, n = tid & 15;
      float z = part_lds[g][0][n] + part_lds[g][1][n] + part_lds[g][2][n] +
                part_lds[g][3][n] + b1_lds[tid];
      z1_lds[tid] = z;
      H1G[h0 + tid] = gelu_fwd(z);
    }
    ++ep;
    batch_barrier(bar, ep * kG);  // h1 complete (also covers W1/W2 of step t-1)

    // ---- P2: z2[slice] = h1 . W2[:,slice]  (WMMA); out, delta2, b2 update ----
    for (int i = tid; i < kH; i += kThreads) h1_lds[i] = H1G[i];
    __syncthreads();
    {
      const int dg = wave & 1;
      const int hq = wave >> 1;
      const int n = lane & 15;
      const int kk = (lane < 16) ? 0 : 2;
      const float* bcol = W2 + (size_t)kk * kD + h0 + dg * 16 + n;
      v8f acc0 = {0.f, 0.f, 0.f, 0.f, 0.f, 0.f, 0.f, 0.f};
      v8f acc1 = acc0;
      const int hbeg = hq * 192;
      for (int hh = 0; hh < 192; hh += 8) {
        const int h = hbeg + hh;
        v2f a0, a1, bb0, bb1;
        a0.x = h1_lds[h + kk];
        a0.y = h1_lds[h + kk + 1];
        a1.x = h1_lds[h + 4 + kk];
        a1.y = h1_lds[h + 4 + kk + 1];
        const float* p0 = bcol + (size_t)h * kD;
        const float* p1 = p0 + (size_t)4 * kD;
        bb0.x = p0[0];
        bb0.y = p0[kD];
        bb1.x = p1[0];
        bb1.y = p1[kD];
        acc0 = wmma_f32_16x16x4(a0, bb0, acc0);
        acc1 = wmma_f32_16x16x4(a1, bb1, acc1);
      }
      v8f acc = acc0 + acc1;
      if (lane < 16) part_lds[dg][hq][n] = acc[0];
    }
    __syncthreads();
    if (tid < kS) {
      const int g = tid >> 4, n = tid & 15;
      float z2 = part_lds[g][0][n] + part_lds[g][1][n] + part_lds[g][2][n] +
                 part_lds[g][3][n] + b2_lds[tid];
      const int d = h0 + tid;
      OUT[((size_t)b * kT + t) * kD + d] = z2;  // v_hat
      const float dl2 = 2.f * (z2 - vt[d]);
      D2G[d] = dl2;
      const float sb = et * Sb2_lds[tid] - th * dl2;
      Sb2_lds[tid] = sb;
      b2_lds[tid] = oma * b2_lds[tid] + sb;
    }
    ++ep;
    batch_barrier(bar, ep * kG);  // delta2 complete

    // ---- Pdh: dh[slice] = delta2 . W2[:,slice] (reference einsum order) ----
    for (int i = tid; i < kD; i += kThreads) d2_lds[i] = D2G[i];
    __syncthreads();
    {
      const int jg = wave & 1;
      const int iq = wave >> 1;
      const int n = lane & 15;
      const int kk = (lane < 16) ? 0 : 2;
      const float* bcol = W2 + (size_t)kk * kD + h0 + jg * 16 + n;
      v8f acc0 = {0.f, 0.f, 0.f, 0.f, 0.f, 0.f, 0.f, 0.f};
      v8f acc1 = acc0;
      const int ibeg = iq * 192;
      for (int ii = 0; ii < 192; ii += 8) {
        const int i = ibeg + ii;
        v2f a0, a1, bb0, bb1;
        a0.x = d2_lds[i + kk];
        a0.y = d2_lds[i + kk + 1];
        a1.x = d2_lds[i + 4 + kk];
        a1.y = d2_lds[i + 4 + kk + 1];
        const float* p0 = bcol + (size_t)i * kD;
        const float* p1 = p0 + (size_t)4 * kD;
        bb0.x = p0[0];
        bb0.y = p0[kD];
        bb1.x = p1[0];
        bb1.y = p1[kD];
        acc0 = wmma_f32_16x16x4(a0, bb0, acc0);
        acc1 = wmma_f32_16x16x4(a1, bb1, acc1);
      }
      v8f acc = acc0 + acc1;
      if (lane < 16) part_lds[jg][iq][n] = acc[0];
    }
    __syncthreads();
    if (tid < kS) {
      const int g = tid >> 4, n = tid & 15;
      const float dh = part_lds[g][0][n] + part_lds[g][1][n] +
                       part_lds[g][2][n] + part_lds[g][3][n];
      dh_lds[tid] = dh;
      const float dl1 = dh * gelu_bwd(z1_lds[tid]);
      dl1_lds[tid] = dl1;
      const float sb = et * Sb1_lds[tid] - th * dl1;
      Sb1_lds[tid] = sb;
      b1_lds[tid] = oma * b1_lds[tid] + sb;
    }
    ++ep;
    batch_barrier(bar, ep * kG);  // all reads of old W2 done -> safe to update

    // ---- P3: W2/SW2 update (own rows), rank-1 grad = h1 (x) delta2 ----
    for (int r = wave; r < kS; r += 8) {
      const float h1v = h1_lds[h0 + r];
      const size_t base = (size_t)(h0 + r) * kD;
      for (int d = lane; d < kD; d += 32) {
        const float g = h1v * d2_lds[d];
        const float sn = et * SW2[base + d] - th * g;
        SW2[base + d] = sn;
        W2[base + d] = oma * W2[base + d] + sn;
      }
    }
    // ---- P5: W1/SW1 update (own cols), rank-1 grad = k (x) delta1 ----
    {
      const float dl1 = dl1_lds[lane];
      for (int d = wave; d < kD; d += 8) {
        const size_t idx = (size_t)d * kH + h0 + lane;
        const float g = k_lds[d] * dl1;
        const float sn = et * SW1[idx] - th * g;
        SW1[idx] = sn;
        W1[idx] = oma * W1[idx] + sn;
      }
    }
    // No barrier needed here: next-step P1 touches only our own W1 columns
    // (in-WG fence at loop top), and W2 readers wait at the next barrier.
  }
}

extern "C" void kernel_launch(void* const* d_in, const int* in_sizes, int n_in,
                              void* d_out, int out_size, void* d_ws,
                              size_t ws_size, hipStream_t stream) {
  (void)in_sizes;
  (void)n_in;
  (void)out_size;
  (void)ws_size;  // requires ~75.6 MB of workspace
  const float* K = (const float*)d_in[0];
  const float* V = (const float*)d_in[1];
  const float* ETA = (const float*)d_in[2];
  const float* ALPHA = (const float*)d_in[3];
  const float* THETA = (const float*)d_in[4];
  const float* W01 = (const float*)d_in[5];
  const float* B01 = (const float*)d_in[6];
  const float* W02 = (const float*)d_in[7];
  const float* B02 = (const float*)d_in[8];

  const size_t MAT = (size_t)kD * kH;
  const size_t per_batch = 4 * MAT + 2 * (size_t)kH;  // floats
  float* ws = (float*)d_ws;
  unsigned* bars = (unsigned*)(ws + (size_t)kB * per_batch);

  // zero the per-batch barrier counters (graph-capture legal: memset node)
  hipMemsetAsync(bars, 0, kB * sizeof(unsigned), stream);

  dim3 grid(kB * kG), block(kThreads);
  memory_mlp_kernel<<<grid, block, 0, stream>>>(K, V, ETA, ALPHA, THETA, W01,
                                                B01, W02, B02, (float*)d_out,
                                                ws, bars);
}